// CPGCN_5712306503711
// MI455X (gfx1250) — compile-verified
//
#include <hip/hip_runtime.h>
#include <hip/hip_bf16.h>

// ---------------------------------------------------------------------------
// CPGCN forward on gfx1250 (MI455X).
//   layer1: xw1 = x @ W1            (50000x512x64, fp32 WMMA 16x16x4)
//           agg1 = dinv^2*xw1 + b1 + scatter-add(norm_e * xw1[row] -> col)
//   layer2: xw2 = relu(agg1) @ W2   (50000x64x64)
//           agg2 = dinv^2*xw2 + b2 + scatter-add
//   heads : pred  = agg2 @ fcW1 + fcb1  -> d_out[0 .. 50000*16)
//           predc = agg2 @ fcW2 + fcb2  -> d_out[50000*16 ..)
// ---------------------------------------------------------------------------

typedef __attribute__((ext_vector_type(2))) float v2f;
typedef __attribute__((ext_vector_type(8))) float v8f;

#define HID 64

static __host__ __device__ inline size_t align64u(size_t v) { return (v + 63) & ~(size_t)63; }

// ---------------- degree / norm kernels ----------------

__global__ void fill_deg_kernel(float* __restrict__ deg, int n) {
    int i = blockIdx.x * blockDim.x + threadIdx.x;
    if (i < n) deg[i] = 1.0f;   // self-loop weight 1 pre-accumulated
}

__global__ void deg_kernel(float* __restrict__ deg, const int* __restrict__ col,
                           const float* __restrict__ w, int E) {
    int e = blockIdx.x * blockDim.x + threadIdx.x;
    if (e < E) unsafeAtomicAdd(&deg[col[e]], w[e]);
}

__global__ void dinv_kernel(float* __restrict__ deg, int n) {
    int i = blockIdx.x * blockDim.x + threadIdx.x;
    if (i < n) {
        float d = deg[i];
        deg[i] = (d > 0.0f) ? rsqrtf(d) : 0.0f;   // in place: deg -> dinv
    }
}

__global__ void norm_kernel(float* __restrict__ norm, const float* __restrict__ dinv,
                            const int* __restrict__ row, const int* __restrict__ col,
                            const float* __restrict__ w, int E) {
    int e = blockIdx.x * blockDim.x + threadIdx.x;
    if (e < E) norm[e] = dinv[row[e]] * w[e] * dinv[col[e]];
}

// agg[i,f] = dinv[i]^2 * xw[i,f] + b[f]   (self-loop term + bias)
__global__ void agg_init_kernel(float* __restrict__ agg, const float* __restrict__ xw,
                                const float* __restrict__ dinv, const float* __restrict__ bias,
                                int n) {
    long long i = (long long)blockIdx.x * blockDim.x + threadIdx.x;
    if (i >= (long long)n * HID) return;
    int node = (int)(i >> 6);
    int f = (int)(i & (HID - 1));
    float d = dinv[node];
    agg[i] = d * d * xw[i] + bias[f];
}

// scatter-add: agg[col,f] += norm[e] * xw[row,f]; 4 feats per thread (float4 gather)
__global__ void scatter_kernel(float* __restrict__ agg, const float* __restrict__ xw,
                               const float* __restrict__ norm, const int* __restrict__ row,
                               const int* __restrict__ col, int E) {
    long long t = (long long)blockIdx.x * blockDim.x + threadIdx.x;
    if (t >= (long long)E * (HID / 4)) return;
    int e = (int)(t >> 4);
    int f = (int)(t & 15) << 2;
    int r = row[e], c = col[e];
    float nv = norm[e];
    const float4 xv = *(const float4*)(xw + (size_t)r * HID + f);
    float* dst = agg + (size_t)c * HID + f;
    unsafeAtomicAdd(dst + 0, nv * xv.x);
    unsafeAtomicAdd(dst + 1, nv * xv.y);
    unsafeAtomicAdd(dst + 2, nv * xv.z);
    unsafeAtomicAdd(dst + 3, nv * xv.w);
}

// ---------------- fp32 WMMA GEMM: C[M,N] = op(A)[M,K] @ B[K,N] (+bias) ----------------
// NT 16-wide N tiles (N = NT*16). 4 waves per block, 1 M-tile (16 rows) per wave.
// B staged through LDS in 64-row K chunks, stored FRAGMENT-MAJOR:
//   element (k,n) at dword  (k>>1)*PSTR + 2*n + (k&1)
// so each lane's B fragment (B[krow][n], B[krow+1][n]) is one aligned ds_load_b64.
// PSTR chosen == 32 (mod 64) dwords -> lane-halves hit disjoint bank halves
// (lanes 0-15: banks 0..31, lanes 16-31: banks 32..63) => conflict-free b64 reads.
// Requires K % 64 == 0, M % 16 == 0.

template <int NT, bool RELU_A, bool BIAS>
__global__ __launch_bounds__(128) void gemm_wmma_kernel(
    const float* __restrict__ A, const float* __restrict__ B,
    const float* __restrict__ bias, float* __restrict__ C, int M, int K) {
    constexpr int N = NT * 16;
    constexpr int PSTR = (((2 * N) & 63) == 32) ? (2 * N) : (2 * N + 32); // dwords per k-pair row
    constexpr int CHUNK = 64;
    __shared__ __align__(16) float ldsB[(CHUNK / 2) * PSTR];

    const int tid = threadIdx.x;
    const int wave = tid >> 5;
    const int lane = tid & 31;
    const int half = lane >> 4;     // K-pair select (A/B) and M+8 select (C/D)
    const int mlane = lane & 15;    // M index (A) / N index (B, C/D)
    const int mtile = blockIdx.x * 4 + wave;
    const int m0 = mtile * 16;
    const bool active = (m0 < M);

    v8f acc[NT];
#pragma unroll
    for (int t = 0; t < NT; ++t)
#pragma unroll
        for (int j = 0; j < 8; ++j) acc[t][j] = 0.0f;

    for (int kc = 0; kc < K; kc += CHUNK) {
        // cooperative fill of B[kc .. kc+63, :] into fragment-major LDS (all threads)
        for (int idx = tid; idx < (CHUNK / 2) * N; idx += 128) {
            int kp = idx / N, c = idx % N;
            v2f bp;
            bp[0] = B[(size_t)(kc + 2 * kp + 0) * N + c];
            bp[1] = B[(size_t)(kc + 2 * kp + 1) * N + c];
            *(v2f*)&ldsB[kp * PSTR + c * 2] = bp;   // ds_store_b64, conflict-free
        }
        __syncthreads();

        if (active) {   // wave-uniform branch: EXEC all-ones inside (WMMA requirement)
            const float* Arow = A + (size_t)(m0 + mlane) * K + kc + half * 2;
#pragma unroll
            for (int kk = 0; kk < CHUNK; kk += 4) {
                float a0 = Arow[kk + 0];
                float a1 = Arow[kk + 1];
                if (RELU_A) { a0 = fmaxf(a0, 0.0f); a1 = fmaxf(a1, 0.0f); }
                v2f af;
                af[0] = a0; af[1] = a1;
                const int kpair = (kk >> 1) + half;   // (kk + 2*half) / 2
#pragma unroll
                for (int nt = 0; nt < NT; ++nt) {
                    const v2f bf = *(const v2f*)&ldsB[kpair * PSTR + (nt * 16 + mlane) * 2];
                    acc[nt] = __builtin_amdgcn_wmma_f32_16x16x4_f32(
                        false, af, false, bf, (short)0, acc[nt], false, false);
                }
            }
        }
        __syncthreads();
    }

    if (active) {
#pragma unroll
        for (int nt = 0; nt < NT; ++nt) {
            const int coln = nt * 16 + mlane;
            float bv = BIAS ? bias[coln] : 0.0f;
#pragma unroll
            for (int j = 0; j < 8; ++j) {
                int rowg = m0 + j + 8 * half;
                C[(size_t)rowg * N + coln] = acc[nt][j] + bv;
            }
        }
    }
}

// ---------------------------------------------------------------------------

static inline unsigned gridFor(long long work, int block) {
    return (unsigned)((work + block - 1) / block);
}

extern "C" void kernel_launch(void* const* d_in, const int* in_sizes, int n_in,
                              void* d_out, int out_size, void* d_ws, size_t ws_size,
                              hipStream_t stream) {
    const float* x    = (const float*)d_in[0];
    const int*   ei   = (const int*)d_in[1];    // [2, E] row-major: row = ei, col = ei+E
    const float* ew   = (const float*)d_in[2];
    const float* W1   = (const float*)d_in[3];
    const float* b1   = (const float*)d_in[4];
    const float* W2   = (const float*)d_in[5];
    const float* b2   = (const float*)d_in[6];
    const float* fcW1 = (const float*)d_in[7];
    const float* fcb1 = (const float*)d_in[8];
    const float* fcW2 = (const float*)d_in[9];
    const float* fcb2 = (const float*)d_in[10];
    float* out = (float*)d_out;

    const int nhid  = in_sizes[4];              // 64
    const int nfeat = in_sizes[3] / nhid;       // 512
    const int n     = in_sizes[0] / nfeat;      // 50000
    const int E     = in_sizes[2];              // 800000
    const int ncls  = in_sizes[8];              // 16
    (void)n_in; (void)out_size; (void)ws_size; (void)nhid;

    const int* erow = ei;
    const int* ecol = ei + E;

    // workspace layout (floats)
    float* ws = (float*)d_ws;
    size_t off = 0;
    float* dinv = ws + off;  off += align64u((size_t)n);
    float* norm = ws + off;  off += align64u((size_t)E);
    float* xw   = ws + off;  off += (size_t)n * HID;
    float* agg  = ws + off;  off += (size_t)n * HID;

    const int T = 256;

    // --- degrees + symmetric norm (shared by both conv layers) ---
    fill_deg_kernel<<<gridFor(n, T), T, 0, stream>>>(dinv, n);
    deg_kernel<<<gridFor(E, T), T, 0, stream>>>(dinv, ecol, ew, E);
    dinv_kernel<<<gridFor(n, T), T, 0, stream>>>(dinv, n);
    norm_kernel<<<gridFor(E, T), T, 0, stream>>>(norm, dinv, erow, ecol, ew, E);

    const unsigned gemmGrid = gridFor((long long)(n / 16), 4);

    // --- layer 1: xw1 = x @ W1 ; agg1 = dinv^2*xw1 + b1 + scatter ---
    gemm_wmma_kernel<4, false, false><<<gemmGrid, 128, 0, stream>>>(x, W1, nullptr, xw, n, nfeat);
    agg_init_kernel<<<gridFor((long long)n * HID, T), T, 0, stream>>>(agg, xw, dinv, b1, n);
    scatter_kernel<<<gridFor((long long)E * (HID / 4), T), T, 0, stream>>>(agg, xw, norm, erow, ecol, E);

    // --- layer 2: xw2 = relu(agg1) @ W2 ; agg2 = dinv^2*xw2 + b2 + scatter ---
    gemm_wmma_kernel<4, true, false><<<gemmGrid, 128, 0, stream>>>(agg, W2, nullptr, xw, n, HID);
    agg_init_kernel<<<gridFor((long long)n * HID, T), T, 0, stream>>>(agg, xw, dinv, b2, n);
    scatter_kernel<<<gridFor((long long)E * (HID / 4), T), T, 0, stream>>>(agg, xw, norm, erow, ecol, E);

    // --- heads: pred = agg2 @ fcW1 + fcb1 ; pred_cluster = agg2 @ fcW2 + fcb2 ---
    gemm_wmma_kernel<1, false, true><<<gemmGrid, 128, 0, stream>>>(agg, fcW1, fcb1, out, n, HID);
    gemm_wmma_kernel<2, false, true><<<gemmGrid, 128, 0, stream>>>(agg, fcW2, fcb2,
                                                                   out + (size_t)n * ncls, n, HID);
}